// APPNP_Net_33217277067912
// MI455X (gfx1250) — compile-verified
//
#include <hip/hip_runtime.h>
#include <math.h>

// ---------------- problem constants (match reference) ----------------
#define N_NODES 100000
#define N_EDGES 1600000
#define IN_C    500
#define HID     256
#define OUT_C   40
#define K_ITERS 10
#define ALPHA   0.1f
#define LN_EPS  1e-5f

typedef __attribute__((ext_vector_type(2))) float v2f;
typedef __attribute__((ext_vector_type(8))) float v8f;

__device__ __forceinline__ float gelu_exact(float v) {
  // exact GELU: 0.5*x*(1+erf(x/sqrt(2)))  (reference uses approximate=False)
  return 0.5f * v * (1.0f + erff(v * 0.70710678118654752f));
}

// ============================================================================
// Fused  Y = LayerNorm(GELU(X @ W + b)) ;  X:[N x Kdim], W:[Kdim x 256]
// One block = 32 rows x all 256 cols. 8 waves: wave (wm,wn) owns a 16-row x
// 64-col strip via 4 fp32 WMMA accumulators.  K is consumed in chunks of 32
// (8 x V_WMMA_F32_16X16X4_F32 steps) staged through LDS.
// ============================================================================
__global__ __launch_bounds__(256)
void mlp_layer_kernel(const float* __restrict__ X, const float* __restrict__ W,
                      const float* __restrict__ bias, const float* __restrict__ gamma,
                      const float* __restrict__ beta, float* __restrict__ Y,
                      int Kdim)
{
  __shared__ float xs[32][36];        // 32 rows x 32 k   (+pad: conflict-free A reads)
  __shared__ float ws_hs[32 * 272];   // K-loop: W tile [32k x 256n]; after: GELU stage
  float (*ws)[272] = (float(*)[272])ws_hs;
  float (*hs)[272] = (float(*)[272])ws_hs;

  const int tid  = threadIdx.x;
  const int lane = tid & 31;
  const int wid  = tid >> 5;
  const int lh   = lane >> 4;    // 0: lanes 0-15, 1: lanes 16-31
  const int ll   = lane & 15;
  const int row0 = blockIdx.x * 32;   // N_NODES % 32 == 0 -> no row guard
  const int wm   = wid >> 2;     // 0..1  (16-row half)
  const int wn   = wid & 3;      // 0..3  (64-col strip)

  v8f acc[4];
#pragma unroll
  for (int t = 0; t < 4; ++t)
#pragma unroll
    for (int i = 0; i < 8; ++i) acc[t][i] = 0.0f;

  const int nch = (Kdim + 31) >> 5;
  for (int kc = 0; kc < nch; ++kc) {
    const int k0 = kc << 5;

    // ---- stage X tile [32 x 32] (rows are 16B aligned: 500*4B and 256*4B) ----
    {
      const int r  = tid >> 3;
      const int kq = (tid & 7) << 2;
      const int gk = k0 + kq;
      const float* src = X + (size_t)(row0 + r) * Kdim + gk;
      if (gk + 3 < Kdim) {
        const float4 v = *(const float4*)src;
        xs[r][kq + 0] = v.x; xs[r][kq + 1] = v.y;
        xs[r][kq + 2] = v.z; xs[r][kq + 3] = v.w;
      } else {
#pragma unroll
        for (int j = 0; j < 4; ++j)
          xs[r][kq + j] = (gk + j < Kdim) ? src[j] : 0.0f;
      }
    }
    // ---- stage W tile [32 x 256], zero-padded past Kdim ----
    {
      const int c4 = (tid & 63) << 2;
      const int kr = tid >> 6;                // 0..3
#pragma unroll
      for (int j = 0; j < 8; ++j) {
        const int kl = (j << 2) + kr;         // 0..31
        const int gk = k0 + kl;
        if (gk < Kdim) {
          const float4 v = *(const float4*)(W + (size_t)gk * HID + c4);
          ws[kl][c4 + 0] = v.x; ws[kl][c4 + 1] = v.y;
          ws[kl][c4 + 2] = v.z; ws[kl][c4 + 3] = v.w;
        } else {
          ws[kl][c4 + 0] = 0.f; ws[kl][c4 + 1] = 0.f;
          ws[kl][c4 + 2] = 0.f; ws[kl][c4 + 3] = 0.f;
        }
      }
    }
    __syncthreads();

    // ---- 8 WMMA steps of K=4 ----
#pragma unroll
    for (int ks = 0; ks < 8; ++ks) {
      const int kb = (ks << 2) + (lh << 1);   // lane-half selects K{0,1} vs K{2,3}
      v2f a;
      a.x = xs[wm * 16 + ll][kb + 0];
      a.y = xs[wm * 16 + ll][kb + 1];
#pragma unroll
      for (int nt = 0; nt < 4; ++nt) {
        const int nb = wn * 64 + nt * 16;
        v2f b;
        b.x = ws[kb + 0][nb + ll];
        b.y = ws[kb + 1][nb + ll];
        acc[nt] = __builtin_amdgcn_wmma_f32_16x16x4_f32(
            false, a, false, b, (short)0, acc[nt], false, false);
      }
    }
    __syncthreads();
  }

  // ---- bias + exact GELU -> LDS stage (C/D layout: VGPR i -> M=i / M=i+8) ----
#pragma unroll
  for (int nt = 0; nt < 4; ++nt) {
    const int c  = wn * 64 + nt * 16 + ll;
    const float bv = bias[c];
#pragma unroll
    for (int i = 0; i < 8; ++i) {
      const int r = wm * 16 + i + (lh << 3);
      hs[r][c] = gelu_exact(acc[nt][i] + bv);
    }
  }
  __syncthreads();

  // ---- LayerNorm over 256 cols: each wave handles 4 rows ----
#pragma unroll
  for (int rr = 0; rr < 4; ++rr) {
    const int r = wid * 4 + rr;
    float v[8], s = 0.f, ss = 0.f;
#pragma unroll
    for (int j = 0; j < 8; ++j) {
      v[j] = hs[r][lane + (j << 5)];
      s  += v[j];
      ss += v[j] * v[j];
    }
#pragma unroll
    for (int off = 16; off; off >>= 1) {
      s  += __shfl_xor(s,  off, 32);
      ss += __shfl_xor(ss, off, 32);
    }
    const float mean = s * (1.0f / 256.0f);
    const float var  = ss * (1.0f / 256.0f) - mean * mean;
    const float rstd = rsqrtf(var + LN_EPS);
    float* yrow = Y + (size_t)(row0 + r) * HID;
#pragma unroll
    for (int j = 0; j < 8; ++j) {
      const int c = lane + (j << 5);
      yrow[c] = (v[j] - mean) * rstd * gamma[c] + beta[c];
    }
  }
}

// ============================================================================
// Output projection:  Y = X @ W3 + b3 ;  X:[N x 256], W3:[256 x 40]
// Cols padded to 64 in LDS (zeros); only c<40 stored.
// ============================================================================
__global__ __launch_bounds__(256)
void out_layer_kernel(const float* __restrict__ X, const float* __restrict__ W,
                      const float* __restrict__ bias, float* __restrict__ Y)
{
  __shared__ float xs[32][36];
  __shared__ float ws[32][80];   // 32k x 64n (pad->80: conflict-free B reads)

  const int tid  = threadIdx.x;
  const int lane = tid & 31;
  const int wid  = tid >> 5;
  const int lh   = lane >> 4;
  const int ll   = lane & 15;
  const int row0 = blockIdx.x * 32;
  const int wm   = wid >> 2;
  const int wn   = wid & 3;      // 16-col tile; wn==3 computes padded zeros

  v8f acc;
#pragma unroll
  for (int i = 0; i < 8; ++i) acc[i] = 0.0f;

  for (int kc = 0; kc < 8; ++kc) {     // Kdim = 256 exactly
    const int k0 = kc << 5;
    {
      const int r  = tid >> 3;
      const int kq = (tid & 7) << 2;
      const float4 v = *(const float4*)(X + (size_t)(row0 + r) * HID + k0 + kq);
      xs[r][kq + 0] = v.x; xs[r][kq + 1] = v.y;
      xs[r][kq + 2] = v.z; xs[r][kq + 3] = v.w;
    }
    {
      const int kl = tid >> 3;               // 0..31
      const int c0 = (tid & 7) << 3;         // 0..56
#pragma unroll
      for (int j = 0; j < 8; ++j) {
        const int c = c0 + j;
        ws[kl][c] = (c < OUT_C) ? W[(size_t)(k0 + kl) * OUT_C + c] : 0.0f;
      }
    }
    __syncthreads();
#pragma unroll
    for (int ks = 0; ks < 8; ++ks) {
      const int kb = (ks << 2) + (lh << 1);
      v2f a, b;
      a.x = xs[wm * 16 + ll][kb + 0];
      a.y = xs[wm * 16 + ll][kb + 1];
      const int nb = wn * 16;
      b.x = ws[kb + 0][nb + ll];
      b.y = ws[kb + 1][nb + ll];
      acc = __builtin_amdgcn_wmma_f32_16x16x4_f32(
          false, a, false, b, (short)0, acc, false, false);
    }
    __syncthreads();
  }

  const int c = wn * 16 + ll;
  if (c < OUT_C) {
    const float bv = bias[c];
#pragma unroll
    for (int i = 0; i < 8; ++i) {
      const int r = row0 + wm * 16 + i + (lh << 3);
      Y[(size_t)r * OUT_C + c] = acc[i] + bv;
    }
  }
}

// ============================================================================
// GCN normalization + CSR build (pull-based propagation: no atomics in steps)
// ============================================================================
__global__ void zero_int_kernel(int* __restrict__ p, int n) {
  const int i = blockIdx.x * blockDim.x + threadIdx.x;
  if (i < n) p[i] = 0;
}

// in-degree (excluding self-loop), integer atomics -> deterministic counts
__global__ void cnt_scatter_kernel(const int* __restrict__ col, int* __restrict__ cnt) {
  const int e = blockIdx.x * blockDim.x + threadIdx.x;
  if (e < N_EDGES) atomicAdd(&cnt[col[e]], 1);
}

// dinv[i] = (deg_with_self_loop)^{-1/2} ; fully deterministic
__global__ void dinv_kernel(const int* __restrict__ cnt, float* __restrict__ dinv) {
  const int i = blockIdx.x * blockDim.x + threadIdx.x;
  if (i < N_NODES) dinv[i] = rsqrtf((float)(cnt[i] + 1));
}

// single-block exclusive prefix scan of cnt -> offs (deterministic)
__global__ __launch_bounds__(1024)
void scan_kernel(const int* __restrict__ cnt, int* __restrict__ offs) {
  __shared__ int sums[1024];
  const int T = 1024;
  const int chunk = (N_NODES + T - 1) / T;            // 98
  const int begin = threadIdx.x * chunk;
  const int end   = (begin + chunk < N_NODES) ? begin + chunk : N_NODES;
  int s = 0;
  for (int i = begin; i < end; ++i) s += cnt[i];
  sums[threadIdx.x] = s;
  __syncthreads();
  for (int off = 1; off < T; off <<= 1) {
    int t = (threadIdx.x >= off) ? sums[threadIdx.x - off] : 0;
    __syncthreads();
    sums[threadIdx.x] += t;
    __syncthreads();
  }
  int run = sums[threadIdx.x] - s;                    // exclusive base of my chunk
  for (int i = begin; i < end; ++i) { offs[i] = run; run += cnt[i]; }
}

// scatter edges into CSR (by target), precompute edge weight dinv[s]*dinv[t]
__global__ void fill_csr_kernel(const int* __restrict__ ei, const int* __restrict__ offs,
                                int* __restrict__ cursor, const float* __restrict__ dinv,
                                int* __restrict__ csr_src, float* __restrict__ csr_w) {
  const int e = blockIdx.x * blockDim.x + threadIdx.x;
  if (e >= N_EDGES) return;
  const int s = ei[e];               // source (row)
  const int t = ei[N_EDGES + e];     // target (col)
  const int p = offs[t] + atomicAdd(&cursor[t], 1);
  csr_src[p] = s;
  csr_w[p]   = dinv[s] * dinv[t];
}

// ============================================================================
// APPNP pull step:  hn[i] = a*h0[i] + (1-a)*( dinv_i^2*hc[i] + sum_j w_ij*hc[src_j] )
// One thread per (node, 4-channel group): idx*16B write -> fully coalesced.
// All gathers are L2-resident (h = 16MB, CSR ~13MB; L2 = 192MB). No atomics.
// ============================================================================
__global__ __launch_bounds__(256)
void appnp_pull_kernel(float* __restrict__ hn, const float* __restrict__ h0,
                       const float* __restrict__ hc, const float* __restrict__ dinv,
                       const int* __restrict__ offs, const int* __restrict__ cnt,
                       const int* __restrict__ csr_src, const float* __restrict__ csr_w) {
  const int idx = blockIdx.x * blockDim.x + threadIdx.x;
  if (idx >= N_NODES * 10) return;
  const int i = idx / 10;
  const int q = idx - i * 10;
  const int b = offs[i];
  const int n = cnt[i];
  const float di = dinv[i];

  const float4 hv = *(const float4*)(hc + (size_t)i * OUT_C + q * 4);
  float4 acc;
  acc.x = di * di * hv.x; acc.y = di * di * hv.y;
  acc.z = di * di * hv.z; acc.w = di * di * hv.w;

  for (int j = 0; j < n; ++j) {
    const int   s = csr_src[b + j];
    const float w = csr_w[b + j];
    if (j + 1 < n)   // gfx1250 global_prefetch_b8 of next gathered row
      __builtin_prefetch(hc + (size_t)csr_src[b + j + 1] * OUT_C + q * 4, 0, 0);
    const float4 v = *(const float4*)(hc + (size_t)s * OUT_C + q * 4);
    acc.x += w * v.x; acc.y += w * v.y; acc.z += w * v.z; acc.w += w * v.w;
  }

  const float4 z = *(const float4*)(h0 + (size_t)i * OUT_C + q * 4);
  float4 o;
  o.x = ALPHA * z.x + (1.0f - ALPHA) * acc.x;
  o.y = ALPHA * z.y + (1.0f - ALPHA) * acc.y;
  o.z = ALPHA * z.z + (1.0f - ALPHA) * acc.z;
  o.w = ALPHA * z.w + (1.0f - ALPHA) * acc.w;
  *(float4*)(hn + (size_t)i * OUT_C + q * 4) = o;
}

// ============================================================================
extern "C" void kernel_launch(void* const* d_in, const int* in_sizes, int n_in,
                              void* d_out, int out_size, void* d_ws, size_t ws_size,
                              hipStream_t stream)
{
  (void)in_sizes; (void)n_in; (void)out_size; (void)ws_size;

  const float* x   = (const float*)d_in[0];
  const int*   ei  = (const int*)d_in[1];    // [2,E] int32 (JAX default x64-off)
  const float* W1  = (const float*)d_in[2];
  const float* b1  = (const float*)d_in[3];
  const float* g1  = (const float*)d_in[4];
  const float* be1 = (const float*)d_in[5];
  const float* W2  = (const float*)d_in[6];
  const float* b2  = (const float*)d_in[7];
  const float* g2  = (const float*)d_in[8];
  const float* be2 = (const float*)d_in[9];
  const float* W3  = (const float*)d_in[10];
  const float* b3  = (const float*)d_in[11];
  float* out = (float*)d_out;

  // ---- workspace carve ----
  // A region (N*HID floats): h1, then ping/pong/dinv after layer 2 consumed it
  // B region (N*HID floats): h2, then CSR arrays after out_layer consumed it
  // C region (N*OUT floats): h0 (persists)
  float* A    = (float*)d_ws;
  float* B    = A + (size_t)N_NODES * HID;
  float* C    = B + (size_t)N_NODES * HID;
  float* P    = A;                                   // ping [N x 40]
  float* Q    = A + (size_t)N_NODES * OUT_C;         // pong [N x 40]
  float* dinv = A + (size_t)2 * N_NODES * OUT_C;     // [N]
  int*   cnt     = (int*)B;                          // [N]
  int*   offs    = cnt + N_NODES;                    // [N]
  int*   cursor  = offs + N_NODES;                   // [N]
  int*   csr_src = cursor + N_NODES;                 // [E]
  float* csr_w   = (float*)(csr_src + N_EDGES);      // [E]

  const dim3 blk(256);
  const int gN   = (N_NODES + 255) / 256;
  const int gE   = (N_EDGES + 255) / 256;

  // ---- MLP (WMMA fp32) ----
  mlp_layer_kernel<<<N_NODES / 32, blk, 0, stream>>>(x, W1, b1, g1, be1, A, IN_C);
  mlp_layer_kernel<<<N_NODES / 32, blk, 0, stream>>>(A, W2, b2, g2, be2, B, HID);
  out_layer_kernel<<<N_NODES / 32, blk, 0, stream>>>(B, W3, b3, C);

  // ---- degree + dinv + CSR build (B region free after out_layer) ----
  zero_int_kernel   <<<gN, blk, 0, stream>>>(cnt, N_NODES);
  cnt_scatter_kernel<<<gE, blk, 0, stream>>>(ei + N_EDGES, cnt);
  dinv_kernel       <<<gN, blk, 0, stream>>>(cnt, dinv);
  scan_kernel       <<<1, 1024, 0, stream>>>(cnt, offs);
  zero_int_kernel   <<<gN, blk, 0, stream>>>(cursor, N_NODES);
  fill_csr_kernel   <<<gE, blk, 0, stream>>>(ei, offs, cursor, dinv, csr_src, csr_w);

  // ---- K APPNP pull steps (atomic-free), last writes d_out ----
  const int gP = (N_NODES * 10 + 255) / 256;
  const float* src = C;
  for (int k = 0; k < K_ITERS; ++k) {
    float* dst = (k == K_ITERS - 1) ? out : ((k & 1) ? Q : P);
    appnp_pull_kernel<<<gP, blk, 0, stream>>>(dst, C, src, dinv, offs, cnt, csr_src, csr_w);
    src = dst;
  }
}